// WLNReactionCenter_75041668595714
// MI455X (gfx1250) — compile-verified
//
#include <hip/hip_runtime.h>
#include <hip/hip_bf16.h>

typedef __attribute__((ext_vector_type(2))) float v2f;
typedef __attribute__((ext_vector_type(8))) float v8f;

#define VNODES 4096
#define NEDGE  32768
#define NEF    262144
#define DN 82
#define DNP 84      // node-feat K padded to multiple of 4 (and even for b64 loads)
#define DE 6
#define DP 11
#define DD 300
#define LD 304      // padded row stride for [V, 300] workspace buffers
#define NT 5

// ---------------------------------------------------------------------------
// f32 WMMA GEMM: C[4096,N] = act( A[4096,K] @ B[K,N] (+bias) (+Cin) )
// Each wave computes a 64x16 strip of C (4 stacked 16x16 tiles) with
// V_WMMA_F32_16X16X4_F32, sharing one B fragment across the 4 WMMAs.
// K-loop is software-pipelined: next iteration's fragments are loaded before
// this iteration's WMMAs so the scheduler can overlap VMEM with matrix ops.
// Requirements: K % 4 == 0 (and K >= 4), lda even, A 8-byte aligned.
// A-frag: a_j = A[m][k0 + 2*half + j]   (one b64 load per subtile)
// B-frag: b_j = B[k0 + 2*half + j][n]   (column clamped to N-1; safe because
//         a WMMA output column depends only on its own B column, and
//         clamped-lane outputs are never stored)
// C:      acc[i] = C[16*tile + i + 8*half][16*ntile + l16]
// ---------------------------------------------------------------------------
__global__ __launch_bounds__(128)
void gemm_f32_wmma(const float* __restrict__ A, int lda,
                   const float* __restrict__ B, int ldb,
                   const float* __restrict__ bias,
                   const float* __restrict__ Cin,
                   float* __restrict__ C, int ldc,
                   int K, int N, int act)
{
    const int wave = threadIdx.x >> 5;
    const int lane = threadIdx.x & 31;
    const int half = lane >> 4;
    const int l16  = lane & 15;
    const int ntile  = blockIdx.y * 4 + wave;
    const int nTiles = (N + 15) >> 4;
    if (ntile >= nTiles) return;               // wave-uniform exit

    const int col  = ntile * 16 + l16;
    const int colL = (col < N) ? col : (N - 1);    // clamped load column
    const int row0 = blockIdx.x * 64 + l16;        // A row, subtile 0

    v8f acc[4] = {};
    const float* a0p = A + (size_t)(row0     ) * lda + 2 * half;
    const float* a1p = A + (size_t)(row0 + 16) * lda + 2 * half;
    const float* a2p = A + (size_t)(row0 + 32) * lda + 2 * half;
    const float* a3p = A + (size_t)(row0 + 48) * lda + 2 * half;
    const float* bp  = B + (size_t)(2 * half) * ldb + colL;
    const size_t bstep = (size_t)4 * (size_t)ldb;  // pointer walk: no per-iter mul

    // prologue: fragments for k0 = 0
    v2f af0 = *(const v2f*)(a0p);
    v2f af1 = *(const v2f*)(a1p);
    v2f af2 = *(const v2f*)(a2p);
    v2f af3 = *(const v2f*)(a3p);
    v2f bf;
    bf.x = bp[0];
    bf.y = bp[ldb];

    for (int k0 = 4; k0 < K; k0 += 4) {
        bp += bstep;
        // prefetch next iteration's fragments (independent of the WMMAs below)
        v2f naf0 = *(const v2f*)(a0p + k0);
        v2f naf1 = *(const v2f*)(a1p + k0);
        v2f naf2 = *(const v2f*)(a2p + k0);
        v2f naf3 = *(const v2f*)(a3p + k0);
        v2f nbf;
        nbf.x = bp[0];
        nbf.y = bp[ldb];
        acc[0] = __builtin_amdgcn_wmma_f32_16x16x4_f32(false, af0, false, bf, (short)0, acc[0], false, false);
        acc[1] = __builtin_amdgcn_wmma_f32_16x16x4_f32(false, af1, false, bf, (short)0, acc[1], false, false);
        acc[2] = __builtin_amdgcn_wmma_f32_16x16x4_f32(false, af2, false, bf, (short)0, acc[2], false, false);
        acc[3] = __builtin_amdgcn_wmma_f32_16x16x4_f32(false, af3, false, bf, (short)0, acc[3], false, false);
        af0 = naf0; af1 = naf1; af2 = naf2; af3 = naf3; bf = nbf;
    }
    // epilogue: last K-step
    acc[0] = __builtin_amdgcn_wmma_f32_16x16x4_f32(false, af0, false, bf, (short)0, acc[0], false, false);
    acc[1] = __builtin_amdgcn_wmma_f32_16x16x4_f32(false, af1, false, bf, (short)0, acc[1], false, false);
    acc[2] = __builtin_amdgcn_wmma_f32_16x16x4_f32(false, af2, false, bf, (short)0, acc[2], false, false);
    acc[3] = __builtin_amdgcn_wmma_f32_16x16x4_f32(false, af3, false, bf, (short)0, acc[3], false, false);

    if (col >= N) return;
    const float bv = bias ? bias[col] : 0.0f;
#pragma unroll
    for (int sub = 0; sub < 4; ++sub) {
#pragma unroll
        for (int i = 0; i < 8; ++i) {
            const int r = blockIdx.x * 64 + sub * 16 + i + 8 * half;
            float v = acc[sub][i] + bv;
            if (Cin) v += Cin[(size_t)r * ldc + col];
            if (act) v = fmaxf(v, 0.0f);
            C[(size_t)r * ldc + col] = v;
        }
    }
}

// ---------------------------------------------------------------------------
__global__ void zero_f32(float* __restrict__ p, int n)
{
    int i = blockIdx.x * blockDim.x + threadIdx.x;
    if (i < n) p[i] = 0.0f;
}

// zero-pad node_feats [V,82] -> [V,84] so the first GEMM has K % 4 == 0
__global__ void pad_nf(const float* __restrict__ nf, float* __restrict__ out)
{
    int i = blockIdx.x * blockDim.x + threadIdx.x;
    if (i >= VNODES * DNP) return;
    int r = i / DNP, c = i - r * DNP;
    out[i] = (c < DN) ? nf[(size_t)r * DN + c] : 0.0f;
}

// m = relu(hWcat[src] + ef@Wcat_bot + b_cat); agg[dst] += m   (fused, atomic)
__global__ void edge_cat_agg(const float* __restrict__ hWcat,
                             const float* __restrict__ ef,
                             const float* __restrict__ Wcat_bot,   // [6,300]
                             const float* __restrict__ b_cat,
                             const int* __restrict__ src,
                             const int* __restrict__ dst,
                             float* __restrict__ agg)
{
    int idx = blockIdx.x * blockDim.x + threadIdx.x;
    if (idx >= NEDGE * DD) return;
    int e = idx / DD, d = idx - e * DD;
    float v = hWcat[(size_t)src[e] * LD + d] + b_cat[d];
    const float* efe = ef + (size_t)e * DE;
#pragma unroll
    for (int p = 0; p < DE; ++p) v += efe[p] * Wcat_bot[p * DD + d];
    v = fmaxf(v, 0.0f);
    atomicAdd(&agg[(size_t)dst[e] * LD + d], v);
}

// msg = msgA[src] * (ef@W_edge_msg); nbr[dst] += msg   (fused, atomic)
__global__ void edge_msg_agg(const float* __restrict__ msgA,
                             const float* __restrict__ ef,
                             const float* __restrict__ W_edge_msg, // [6,300]
                             const int* __restrict__ src,
                             const int* __restrict__ dst,
                             float* __restrict__ nbr)
{
    int idx = blockIdx.x * blockDim.x + threadIdx.x;
    if (idx >= NEDGE * DD) return;
    int e = idx / DD, d = idx - e * DD;
    const float* efe = ef + (size_t)e * DE;
    float em = 0.0f;
#pragma unroll
    for (int p = 0; p < DE; ++p) em += efe[p] * W_edge_msg[p * DD + d];
    float v = msgA[(size_t)src[e] * LD + d] * em;
    atomicAdd(&nbr[(size_t)dst[e] * LD + d], v);
}

__global__ void mul_vd(const float* __restrict__ a, const float* __restrict__ b,
                       float* __restrict__ c, int n)
{
    int i = blockIdx.x * blockDim.x + threadIdx.x;
    if (i < n) c[i] = a[i] * b[i];
}

// a[e] = sigmoid( sum_d relu(hWc[s,d]+hWc[t,d]+pair[e]@Wc_pair+bc_pair[d]) * Wc_att[d] + bc_att )
__global__ void attn_kernel(const float* __restrict__ hWc,
                            const float* __restrict__ pair,
                            const float* __restrict__ Wc_pair,     // [11,300]
                            const float* __restrict__ bc_pair,
                            const float* __restrict__ Wc_att,      // [300,1]
                            const float* __restrict__ bc_att,
                            const int* __restrict__ csrc,
                            const int* __restrict__ cdst,
                            float* __restrict__ a_out)
{
    int e = blockIdx.x * blockDim.x + threadIdx.x;
    if (e >= NEF) return;
    const int s = csrc[e], t = cdst[e];
    float pf[DP];
#pragma unroll
    for (int p = 0; p < DP; ++p) pf[p] = pair[(size_t)e * DP + p];
    const float* hs = hWc + (size_t)s * LD;
    const float* ht = hWc + (size_t)t * LD;
    float acc = bc_att[0];
    for (int d = 0; d < DD; ++d) {
        float x = hs[d] + ht[d] + bc_pair[d];
#pragma unroll
        for (int p = 0; p < DP; ++p) x += pf[p] * Wc_pair[p * DD + d];
        x = fmaxf(x, 0.0f);
        acc += x * Wc_att[d];
    }
    a_out[e] = 1.0f / (1.0f + __expf(-acc));
}

// context[b*64+i, d] = sum_j hF[b*64+j, d] * a[b*4096 + j*64 + i]
// One workgroup per molecule: the 64x64 attention block (16 KB) is staged in
// LDS once, turning the 256B-strided gather into LDS reads; hF loads are
// lane-coalesced (consecutive threads -> consecutive d).
__global__ __launch_bounds__(256)
void context_kernel(const float* __restrict__ hF,
                    const float* __restrict__ a,
                    float* __restrict__ ctx)
{
    __shared__ float as[64 * 64];
    const int b = blockIdx.x;
    const float* ab = a + (size_t)b * 4096;
    for (int i = threadIdx.x; i < 4096; i += 256) as[i] = ab[i];
    __syncthreads();
    for (int o = threadIdx.x; o < 64 * DD; o += 256) {
        const int i = o / DD, d = o - i * DD;
        const float* hb = hF + (size_t)(b * 64) * LD + d;
        float s = 0.0f;
#pragma unroll 8
        for (int j = 0; j < 64; ++j) s += hb[(size_t)j * LD] * as[j * 64 + i];
        ctx[(size_t)(b * 64 + i) * LD + d] = s;
    }
}

// pre[d] = hpc[s,d]+hpc[t,d]+pair[e]@Wp_pair+bp_ctx[d];
// scores[e,k] = relu(pre)@Wp_out + bp_out; out2 = scores - 10000*[s==t]
__global__ void score_kernel(const float* __restrict__ hpc,
                             const float* __restrict__ pair,
                             const float* __restrict__ Wp_pair,    // [11,300]
                             const float* __restrict__ bp_ctx,
                             const float* __restrict__ Wp_out,     // [300,5]
                             const float* __restrict__ bp_out,
                             const int* __restrict__ csrc,
                             const int* __restrict__ cdst,
                             float* __restrict__ out)
{
    int e = blockIdx.x * blockDim.x + threadIdx.x;
    if (e >= NEF) return;
    const int s = csrc[e], t = cdst[e];
    float pf[DP];
#pragma unroll
    for (int p = 0; p < DP; ++p) pf[p] = pair[(size_t)e * DP + p];
    const float* hs = hpc + (size_t)s * LD;
    const float* ht = hpc + (size_t)t * LD;
    float acc[NT] = {0.0f, 0.0f, 0.0f, 0.0f, 0.0f};
    for (int d = 0; d < DD; ++d) {
        float x = hs[d] + ht[d] + bp_ctx[d];
#pragma unroll
        for (int p = 0; p < DP; ++p) x += pf[p] * Wp_pair[p * DD + d];
        x = fmaxf(x, 0.0f);
#pragma unroll
        for (int k = 0; k < NT; ++k) acc[k] += x * Wp_out[d * NT + k];
    }
    const float bias = (s == t) ? 10000.0f : 0.0f;
#pragma unroll
    for (int k = 0; k < NT; ++k) {
        float v = acc[k] + bp_out[k];
        out[(size_t)e * NT + k] = v;
        out[(size_t)NEF * NT + (size_t)e * NT + k] = v - bias;
    }
}

// ---------------------------------------------------------------------------
extern "C" void kernel_launch(void* const* d_in, const int* in_sizes, int n_in,
                              void* d_out, int out_size, void* d_ws, size_t ws_size,
                              hipStream_t stream)
{
    const float* node_feats = (const float*)d_in[0];
    const float* edge_feats = (const float*)d_in[1];
    const float* pair_feats = (const float*)d_in[2];
    const int*   mol_src    = (const int*)d_in[3];
    const int*   mol_dst    = (const int*)d_in[4];
    const int*   comp_src   = (const int*)d_in[5];
    const int*   comp_dst   = (const int*)d_in[6];
    const float* W_in       = (const float*)d_in[7];
    const float* b_in       = (const float*)d_in[8];
    const float* W_cat      = (const float*)d_in[9];   // [306,300]
    const float* b_cat      = (const float*)d_in[10];
    const float* W_new      = (const float*)d_in[11];  // [600,300]
    const float* b_new      = (const float*)d_in[12];
    const float* W_edge_msg = (const float*)d_in[13];
    const float* W_node_msg = (const float*)d_in[14];
    const float* W_self     = (const float*)d_in[15];
    const float* Wc_fsum    = (const float*)d_in[16];
    const float* Wc_pair    = (const float*)d_in[17];
    const float* bc_pair    = (const float*)d_in[18];
    const float* Wc_att     = (const float*)d_in[19];
    const float* bc_att     = (const float*)d_in[20];
    const float* Wp_fsum    = (const float*)d_in[21];
    const float* Wp_pair    = (const float*)d_in[22];
    const float* Wp_ctx     = (const float*)d_in[23];
    const float* bp_ctx     = (const float*)d_in[24];
    const float* Wp_out     = (const float*)d_in[25];
    const float* bp_out     = (const float*)d_in[26];

    float* ws = (float*)d_ws;
    const size_t VB = (size_t)VNODES * LD;   // 1,245,184 floats per node-buffer
    float* hA    = ws + 0 * VB;
    float* hB    = ws + 1 * VB;
    float* tA    = ws + 2 * VB;
    float* tB    = ws + 3 * VB;
    float* agg   = ws + 4 * VB;
    float* hWc   = ws + 5 * VB;
    float* hpc   = ws + 6 * VB;
    float* ctx   = ws + 7 * VB;
    float* a_att = ws + 8 * VB;                       // NEF floats
    float* nfp   = ws + 8 * VB + (size_t)NEF;         // [V, 84] padded node feats

    const dim3 gblk(128);
    const dim3 ggrd(VNODES / 64, (((DD + 15) / 16) + 3) / 4);   // (64, 5)
    auto gemm = [&](const float* A, int lda, const float* B, int ldb,
                    const float* bias, const float* Cin, float* C, int K, int act) {
        gemm_f32_wmma<<<ggrd, gblk, 0, stream>>>(A, lda, B, ldb, bias, Cin, C,
                                                 LD, K, DD, act);
    };

    const int ZB  = (int)((VB + 255) / 256);
    const int EGB = (NEDGE * DD + 255) / 256;     // 38400 blocks
    const int EFB = (NEF + 255) / 256;            // 1024 blocks
    const int PNB = (VNODES * DNP + 255) / 256;

    // h = relu(node_feats_padded @ W_in + b_in)
    pad_nf<<<PNB, 256, 0, stream>>>(node_feats, nfp);
    gemm(nfp, DNP, W_in, DD, b_in, nullptr, hA, DNP, 1);   // cols 82..83 of A are 0

    float* h  = hA;
    float* hN = hB;
    for (int l = 0; l < 3; ++l) {
        // hWcat = h @ W_cat[0:300,:]
        gemm(h, LD, W_cat, DD, nullptr, nullptr, tA, DD, 0);
        zero_f32<<<ZB, 256, 0, stream>>>(agg, (int)VB);
        edge_cat_agg<<<EGB, 256, 0, stream>>>(tA, edge_feats, W_cat + DD * DD,
                                              b_cat, mol_src, mol_dst, agg);
        // h_new = relu(h @ Wn_top + agg @ Wn_bot + b_new)
        gemm(h,   LD, W_new,           DD, b_new,   nullptr, tB, DD, 0);
        gemm(agg, LD, W_new + DD * DD, DD, nullptr, tB,      hN, DD, 1);
        float* sw = h; h = hN; hN = sw;
    }

    // set-comparison readout: h = (h@W_self) * segsum((h@W_node_msg)[src] * ef@W_edge_msg)
    gemm(h, LD, W_node_msg, DD, nullptr, nullptr, tA, DD, 0);
    zero_f32<<<ZB, 256, 0, stream>>>(agg, (int)VB);
    edge_msg_agg<<<EGB, 256, 0, stream>>>(tA, edge_feats, W_edge_msg,
                                          mol_src, mol_dst, agg);
    gemm(h, LD, W_self, DD, nullptr, nullptr, tB, DD, 0);
    mul_vd<<<ZB, 256, 0, stream>>>(tB, agg, hN, (int)VB);
    float* hF = hN;                      // final node representation

    // factored pairwise GEMMs: feat_sum@W == (h@W)[s] + (h@W)[t]
    gemm(hF, LD, Wc_fsum, DD, nullptr, nullptr, hWc, DD, 0);
    gemm(hF, LD, Wp_fsum, DD, nullptr, nullptr, hpc, DD, 0);   // hpc := h@Wp_fsum

    attn_kernel<<<EFB, 256, 0, stream>>>(hWc, pair_feats, Wc_pair, bc_pair,
                                         Wc_att, bc_att, comp_src, comp_dst, a_att);
    context_kernel<<<64, 256, 0, stream>>>(hF, a_att, ctx);

    // hpc += ctx @ Wp_ctx   (ctx_sum@Wp_ctx == ctxW[s]+ctxW[t])
    gemm(ctx, LD, Wp_ctx, DD, nullptr, hpc, hpc, DD, 0);

    score_kernel<<<EFB, 256, 0, stream>>>(hpc, pair_feats, Wp_pair, bp_ctx,
                                          Wp_out, bp_out, comp_src, comp_dst,
                                          (float*)d_out);
}